// GCN_layer_541165879956
// MI455X (gfx1250) — compile-verified
//
#include <hip/hip_runtime.h>
#include <math.h>

// GCN layer: out = D^{-1/2} A D^{-1/2} X   (COO, E=6.4M, N=100K, D=32)
// Memory/atomic-bound (AI ~ 0.4 FLOP/B). Feature table (12.8MB) fits in L2
// (192MB) so random gathers are L2-resident. wave32 == D: lane-per-dim gives
// fully coalesced 128B gathers and per-cacheline-coalesced f32 atomics.

#define DFEAT 32
#define TILE  256

// ---------------- CDNA5 async global -> LDS staging ----------------
#if defined(__gfx1250__) && __has_builtin(__builtin_amdgcn_global_load_async_to_lds_b32)
#define HAVE_ASYNC_LDS 1
#endif

#ifdef HAVE_ASYNC_LDS
typedef __attribute__((address_space(1))) int gint_t;   // global int*
typedef __attribute__((address_space(3))) int lint_t;   // LDS int*
__device__ __forceinline__ void async_g2l_b32(const void* g, void* l) {
  // signature: (int AS1*, int AS3*, imm offset, imm cpol)
  int* gp = (int*)const_cast<void*>(g);
  int* lp = (int*)l;
  __builtin_amdgcn_global_load_async_to_lds_b32(
      (gint_t*)gp, (lint_t*)lp, 0, 0);
}
#endif

__device__ __forceinline__ void atom_add_f32(float* p, float v) {
  // hardware global_atomic_add_f32 (coarse-grained memory)
  unsafeAtomicAdd(p, v);
}

// ---------------- Phase 1: degree = segment_sum(vals, rows) ----------------
extern "C" __global__ void k_deg(const int* __restrict__ rows,
                                 const float* __restrict__ vals,
                                 float* __restrict__ deg, int E) {
  int e = blockIdx.x * blockDim.x + threadIdx.x;
  if (e < E) {
    __builtin_prefetch(rows + e + 16384, 0, 0);   // global_prefetch_b8 (speculative)
    int   r = __builtin_nontemporal_load(rows + e);
    float v = __builtin_nontemporal_load(vals + e);
    atom_add_f32(&deg[r], v);
  }
}

// ---------------- Phase 2: d^{-1/2} ----------------
extern "C" __global__ void k_dinv(const float* __restrict__ deg,
                                  float* __restrict__ dinv, int N) {
  int i = blockIdx.x * blockDim.x + threadIdx.x;
  if (i < N) {
    float r = deg[i];
    dinv[i] = (r > 0.0f) ? (1.0f / sqrtf(r)) : 0.0f;
  }
}

// ---------------- Phase 3: SpMM with async-staged edge tiles ----------------
extern "C" __global__ void __launch_bounds__(256)
k_spmm(const int* __restrict__ rows, const int* __restrict__ cols,
       const float* __restrict__ vals, const float* __restrict__ dinv,
       const float* __restrict__ feat, float* __restrict__ accum, int E) {
  __shared__ int   s_row[TILE];
  __shared__ int   s_col[TILE];
  __shared__ float s_w[TILE];

  const int t    = threadIdx.x;
  const int lane = t & 31;         // feature dim (wave32 == D)
  const int wv   = t >> 5;         // wave id: owns edge slots [wv*32, wv*32+32)
  const int base = blockIdx.x * TILE;
  const int rem  = E - base;
  const int nTile = rem < TILE ? rem : TILE;

  int e = base + t;
  if (e >= E) e = E - 1;           // clamp: valid address, result masked below

#ifdef HAVE_ASYNC_LDS
  async_g2l_b32(rows + e, &s_row[t]);
  async_g2l_b32(cols + e, &s_col[t]);
  async_g2l_b32(vals + e, &s_w[t]);
  asm volatile("s_wait_asynccnt 0" ::: "memory");
#else
  s_row[t] = __builtin_nontemporal_load(rows + e);
  s_col[t] = __builtin_nontemporal_load(cols + e);
  s_w[t]   = __builtin_nontemporal_load(vals + e);
#endif

  // per-edge normalized weight (own slot only; no cross-thread hazard yet)
  float w = dinv[s_row[t]] * s_w[t] * dinv[s_col[t]];
  s_w[t] = w;
  __syncthreads();

  // each wave sweeps its 32 edges; lane = dim -> 128B coalesced gather + atomic
  const int kbase = wv << 5;
#pragma unroll 4
  for (int j = 0; j < 32; ++j) {
    int k = kbase + j;
    if (k < nTile) {
      int   r  = s_row[k];               // LDS broadcast reads
      int   c  = s_col[k];
      float wt = s_w[k];
      float x  = feat[c * DFEAT + lane]; // L2-resident gather, coalesced
      atom_add_f32(&accum[r * DFEAT + lane], wt * x);
    }
  }
}

// ---------------- Phase 4: output assembly ----------------
extern "C" __global__ void k_copy4(const float4* __restrict__ src,
                                   float4* __restrict__ dst, int n4) {
  int i = blockIdx.x * blockDim.x + threadIdx.x;
  if (i < n4) dst[i] = src[i];
}

// new[index[i], :] = accum[i, :]   (8 float4 per row)
extern "C" __global__ void k_scatter(const int* __restrict__ index,
                                     const float4* __restrict__ accum4,
                                     float4* __restrict__ out4, int N) {
  int idx = blockIdx.x * blockDim.x + threadIdx.x;
  if (idx < N * 8) {
    int i = idx >> 3, q = idx & 7;
    int r = index[i];
    out4[r * 8 + q] = accum4[i * 8 + q];
  }
}

// fallback path: mark rows covered by index, restore features elsewhere
extern "C" __global__ void k_mask(const int* __restrict__ index,
                                  int* __restrict__ mask, int N) {
  int i = blockIdx.x * blockDim.x + threadIdx.x;
  if (i < N) mask[index[i]] = 1;
}

extern "C" __global__ void k_fill(const int* __restrict__ mask,
                                  const float4* __restrict__ feat4,
                                  float4* __restrict__ out4, int N) {
  int idx = blockIdx.x * blockDim.x + threadIdx.x;
  if (idx < N * 8) {
    int r = idx >> 3;
    if (!mask[r]) out4[idx] = feat4[idx];
  }
}

// ---------------- host launcher ----------------
extern "C" void kernel_launch(void* const* d_in, const int* in_sizes, int n_in,
                              void* d_out, int out_size, void* d_ws, size_t ws_size,
                              hipStream_t stream) {
  const float* feat  = (const float*)d_in[0];
  const int*   rows  = (const int*)d_in[1];
  const int*   cols  = (const int*)d_in[2];
  const float* vals  = (const float*)d_in[3];
  const int*   index = (const int*)d_in[4];

  const int E  = in_sizes[1];
  const int ND = in_sizes[0];
  const int N  = ND / DFEAT;
  float* outp  = (float*)d_out;

  float* deg  = (float*)d_ws;
  float* dinv = deg + N;

  const int B = 256;
  const int gE    = (E + B - 1) / B;
  const int gN    = (N + B - 1) / B;
  const int gTile = (E + TILE - 1) / TILE;
  const int n4    = ND / 4;
  const int g4    = (n4 + B - 1) / B;
  const int gRow4 = (N * 8 + B - 1) / B;

  const size_t need_full = (size_t)(2 * N + ND) * sizeof(float);

  (void)hipMemsetAsync(deg, 0, (size_t)N * sizeof(float), stream);

  if (ws_size >= need_full) {
    // general path: accumulate into scratch, then features-copy + index-scatter
    float* accum = dinv + N;
    (void)hipMemsetAsync(accum, 0, (size_t)ND * sizeof(float), stream);
    k_deg <<<gE, B, 0, stream>>>(rows, vals, deg, E);
    k_dinv<<<gN, B, 0, stream>>>(deg, dinv, N);
    k_spmm<<<gTile, 256, 0, stream>>>(rows, cols, vals, dinv, feat, accum, E);
    k_copy4  <<<g4, B, 0, stream>>>((const float4*)feat, (float4*)outp, n4);
    k_scatter<<<gRow4, B, 0, stream>>>(index, (const float4*)accum,
                                       (float4*)outp, N);
  } else {
    // small-ws path: accumulate directly into d_out, mask-restore uncovered rows
    int* mask = (int*)(dinv + N);
    (void)hipMemsetAsync(mask, 0, (size_t)N * sizeof(int), stream);
    (void)hipMemsetAsync(outp, 0, (size_t)ND * sizeof(float), stream);
    k_deg <<<gE, B, 0, stream>>>(rows, vals, deg, E);
    k_dinv<<<gN, B, 0, stream>>>(deg, dinv, N);
    k_mask<<<gN, B, 0, stream>>>(index, mask, N);
    k_spmm<<<gTile, 256, 0, stream>>>(rows, cols, vals, dinv, feat, outp, E);
    k_fill<<<gRow4, B, 0, stream>>>(mask, (const float4*)feat,
                                    (float4*)outp, N);
  }
}